// NMS_81922206204546
// MI455X (gfx1250) — compile-verified
//
#include <hip/hip_runtime.h>
#include <stdint.h>

// ---------------- problem constants ----------------
#define NB     8
#define HH     1024
#define WW     1024
#define R      25
#define KWIN   51
#define TILE   32
#define HALO   (TILE + 2 * R)   // 82
#define LDW    (HALO + 2)       // 84 (padded row stride to dodge bank conflicts)
#define MAXD   4096
#define NPIX   (NB * HH * WW)   // 8388608
#define NWORDS (NPIX / 32)      // 262144
#define CBLOCKS 256
#define WPB    (NWORDS / CBLOCKS) // 1024 words per compaction block
#define WPT    (WPB / 256)        // 4 words per thread (contiguous, order-preserving)

// ---------------- CDNA5 feature probes ----------------
#if defined(__has_builtin)
#if __has_builtin(__builtin_amdgcn_global_load_async_to_lds_b32)
#define HAVE_ASYNC_LDS 1
#endif
#if __has_builtin(__builtin_amdgcn_s_wait_asynccnt)
#define HAVE_WAIT_ASYNC 1
#endif
#if __has_builtin(__builtin_amdgcn_ballot_w32)
#define HAVE_BALLOT32 1
#endif
#endif

__device__ __forceinline__ unsigned ballot32(bool p) {
#ifdef HAVE_BALLOT32
  return __builtin_amdgcn_ballot_w32(p);
#else
  return (unsigned)__ballot((int)p);
#endif
}

__device__ __forceinline__ void wait_async_lds() {
#ifdef HAVE_ASYNC_LDS
#ifdef HAVE_WAIT_ASYNC
  __builtin_amdgcn_s_wait_asynccnt(0);
#else
  asm volatile("s_wait_asynccnt 0" ::: "memory");
#endif
#endif
}

// =====================================================================
// Kernel 1: fused separable 51x51 max-pool + peak mask -> bit words
//   pool via two overlapping 32-length running maxes (5 doubling passes)
// =====================================================================
__global__ __launch_bounds__(256) void pool_mask_kernel(
    const float* __restrict__ x, unsigned* __restrict__ maskbits) {
  __shared__ float lds[2 * HALO * LDW];  // 55,104 B
  float* A = lds;
  float* B = lds + HALO * LDW;

  const int bid = blockIdx.x;
  const int tx  = bid & (WW / TILE - 1);
  const int ty  = (bid / (WW / TILE)) & (HH / TILE - 1);
  const int n   = bid / ((WW / TILE) * (HH / TILE));
  const int x0  = tx * TILE;
  const int y0  = ty * TILE;
  const int t   = threadIdx.x;

  const float NEG = -__builtin_inff();
  const float* __restrict__ xb = x + (size_t)n * HH * WW;

  // ---- pre-fill with -inf so out-of-image halo is neutral for max ----
  for (int e = t; e < HALO * LDW; e += 256) A[e] = NEG;
  __syncthreads();

  // ---- stage 82x82 haloed tile into LDS (async global->LDS on CDNA5) ----
  for (int e = t; e < HALO * HALO; e += 256) {
    const int r  = e / HALO;
    const int c  = e - r * HALO;
    const int gy = y0 - R + r;
    const int gx = x0 - R + c;
    if ((unsigned)gy < (unsigned)HH && (unsigned)gx < (unsigned)WW) {
#ifdef HAVE_ASYNC_LDS
      __builtin_amdgcn_global_load_async_to_lds_b32(
          (int*)(xb + (size_t)gy * WW + gx),
          (int*)(&A[r * LDW + c]), 0, 0);
#else
      A[r * LDW + c] = xb[(size_t)gy * WW + gx];
#endif
    }
  }
  wait_async_lds();
  __syncthreads();

  // ---- horizontal doubling passes: m2,m4,m8,m16,m32 (ping-pong A<->B) ----
  {
    float* src = A;
    float* dst = B;
#pragma unroll
    for (int k = 0; k < 5; ++k) {
      const int off = 1 << k;
      for (int e = t; e < HALO * HALO; e += 256) {
        const int r  = e / HALO;
        const int c  = e - r * HALO;
        int c2 = c + off;
        if (c2 > HALO - 1) c2 = HALO - 1;  // clamp: edge garbage unused
        dst[r * LDW + c] = fmaxf(src[r * LDW + c], src[r * LDW + c2]);
      }
      __syncthreads();
      float* tmp = src; src = dst; dst = tmp;
    }
    // after 5 swaps: m32 lives in B; A region is free for reuse
  }

  // ---- combine two 32-windows -> 51-wide hmax (82 rows x 32 cols) ----
  float* C = A;               // 82*32 floats
  float* D = A + HALO * TILE; // 82*32 floats (both fit inside A's region)
  for (int e = t; e < HALO * TILE; e += 256) {
    const int r  = e / TILE;
    const int wl = e - r * TILE;
    C[e] = fmaxf(B[r * LDW + wl], B[r * LDW + wl + (KWIN - 32)]); // +19
  }
  __syncthreads();

  // ---- vertical doubling passes on hmax (ping-pong C<->D) ----
  float* vs = C;
  float* vd = D;
#pragma unroll
  for (int k = 0; k < 5; ++k) {
    const int off = 1 << k;
    for (int e = t; e < HALO * TILE; e += 256) {
      const int r  = e / TILE;
      const int wl = e - r * TILE;
      int r2 = r + off;
      if (r2 > HALO - 1) r2 = HALO - 1;
      vd[e] = fmaxf(vs[e], vs[r2 * TILE + wl]);
    }
    __syncthreads();
    float* tmp = vs; vs = vd; vd = tmp;
  }
  // v32 lives in vs (82 rows valid up to row 50)

  // ---- peak mask + wave32 ballot -> one 32-bit word per tile row ----
  const int lane = t & 31;
  const int wv   = t >> 5;  // 8 waves, 4 rows each
  for (int rr = wv; rr < TILE; rr += 8) {
    const float pool = fmaxf(vs[rr * TILE + lane],
                             vs[(rr + (KWIN - 32)) * TILE + lane]);
    const int gy = y0 + rr;
    const int gx = x0 + lane;
    const float xv = xb[(size_t)gy * WW + gx];
    const bool m = (xv == pool) && (pool > 0.5f);
    const unsigned bits = ballot32(m);
    if (lane == 0) {
      const size_t p = ((size_t)n * HH + gy) * WW + x0;
      maskbits[p >> 5] = bits;
    }
  }
}

// =====================================================================
// Kernel 2: per-block popcounts of mask words
// =====================================================================
__global__ __launch_bounds__(256) void count_kernel(
    const unsigned* __restrict__ maskbits, unsigned* __restrict__ blockSums) {
  __shared__ unsigned s[256];
  const int t = threadIdx.x;
  const int b = blockIdx.x;
  unsigned cnt = 0;
  const int base = b * WPB;
  for (int i = t; i < WPB; i += 256) cnt += (unsigned)__popc(maskbits[base + i]);
  s[t] = cnt;
  __syncthreads();
#pragma unroll
  for (int st = 128; st > 0; st >>= 1) {
    if (t < st) s[t] += s[t + st];
    __syncthreads();
  }
  if (t == 0) blockSums[b] = s[0];
}

// =====================================================================
// Kernel 3: single-block exclusive scan of 256 block sums
// =====================================================================
__global__ __launch_bounds__(256) void scan_kernel(
    const unsigned* __restrict__ blockSums, unsigned* __restrict__ blockOffs) {
  __shared__ unsigned s[256];
  const int t = threadIdx.x;
  s[t] = blockSums[t];
  __syncthreads();
#pragma unroll
  for (int off = 1; off < 256; off <<= 1) {
    unsigned add = (t >= off) ? s[t - off] : 0u;
    __syncthreads();
    s[t] += add;
    __syncthreads();
  }
  blockOffs[t] = (t == 0) ? 0u : s[t - 1];
}

// =====================================================================
// Kernel 4: fill output with -1
// =====================================================================
__global__ __launch_bounds__(256) void init_kernel(int* __restrict__ out) {
  const int i = blockIdx.x * 256 + threadIdx.x;
  if (i < MAXD * 4) out[i] = -1;
}

// =====================================================================
// Kernel 5: order-preserving scatter of detection coordinates
// =====================================================================
__global__ __launch_bounds__(256) void scatter_kernel(
    const unsigned* __restrict__ maskbits, const unsigned* __restrict__ blockOffs,
    int* __restrict__ out) {
  __shared__ unsigned s[256];
  const int t = threadIdx.x;
  const int b = blockIdx.x;
  const int wbase = b * WPB + t * WPT;  // contiguous words per thread -> order kept

  unsigned w0[WPT];
  unsigned cnt = 0;
#pragma unroll
  for (int i = 0; i < WPT; ++i) {
    w0[i] = maskbits[wbase + i];
    cnt += (unsigned)__popc(w0[i]);
  }
  s[t] = cnt;
  __syncthreads();
  // inclusive scan of per-thread counts
#pragma unroll
  for (int off = 1; off < 256; off <<= 1) {
    unsigned add = (t >= off) ? s[t - off] : 0u;
    __syncthreads();
    s[t] += add;
    __syncthreads();
  }
  unsigned rank = blockOffs[b] + s[t] - cnt;  // exclusive base for this thread

#pragma unroll
  for (int i = 0; i < WPT; ++i) {
    unsigned word = w0[i];
    const size_t pbase = ((size_t)(wbase + i)) << 5;
    while (word) {
      const int bit = __ffs(word) - 1;
      word &= word - 1;
      if (rank < MAXD) {
        const size_t p = pbase + (size_t)bit;
        const int n = (int)(p >> 20);
        const int h = (int)((p >> 10) & 1023u);
        const int w = (int)(p & 1023u);
        int* row = out + (size_t)rank * 4;
        row[0] = n;
        row[1] = 0;
        row[2] = h;
        row[3] = w;
      }
      ++rank;
    }
  }
}

// =====================================================================
extern "C" void kernel_launch(void* const* d_in, const int* in_sizes, int n_in,
                              void* d_out, int out_size, void* d_ws, size_t ws_size,
                              hipStream_t stream) {
  const float* x = (const float*)d_in[0];
  int* out = (int*)d_out;

  // workspace layout
  unsigned* maskbits  = (unsigned*)d_ws;                      // NWORDS * 4 = 1 MB
  unsigned* blockSums = maskbits + NWORDS;                    // 256 * 4
  unsigned* blockOffs = blockSums + CBLOCKS;                  // 256 * 4

  const int ntiles = NB * (HH / TILE) * (WW / TILE);          // 8192

  pool_mask_kernel<<<ntiles, 256, 0, stream>>>(x, maskbits);
  count_kernel<<<CBLOCKS, 256, 0, stream>>>(maskbits, blockSums);
  scan_kernel<<<1, 256, 0, stream>>>(blockSums, blockOffs);
  init_kernel<<<(MAXD * 4 + 255) / 256, 256, 0, stream>>>(out);
  scatter_kernel<<<CBLOCKS, 256, 0, stream>>>(maskbits, blockOffs, out);

  (void)in_sizes; (void)n_in; (void)out_size; (void)ws_size;
}